// NetTransform_88089779241195
// MI455X (gfx1250) — compile-verified
//
#include <hip/hip_runtime.h>
#include <math.h>

// Problem constants (from reference): B=8, TN=1, T=1024, H=512, D=509
#define TDIM 1024
#define HDIM 512
#define DDIM 509
#define BSZ  8
#define NT   64                       // TDIM / 16
#define TILES_PER_B (NT * (NT + 1) / 2)  // 2080 lower-tri 16x16 tiles

typedef __attribute__((ext_vector_type(2))) float v2f;
typedef __attribute__((ext_vector_type(8))) float v8f;

// Faithful sequential band quantization from the reference
__device__ __forceinline__ float energy_quant(float x) {
    const float l0 = 13.f/32.f, l1 = 8.f/32.f, l2 = 5.f/32.f,
                l3 = 3.f/32.f,  l4 = 2.f/32.f, l5 = 1.f/32.f;
    float y = (x > l0) ? l0 : x;
    y = (l0 > y && y > l1) ? l1 : y;
    y = (l1 > y && y > l2) ? l2 : y;
    y = (l2 > y && y > l3) ? l3 : y;
    y = (l3 > y && y > l4) ? l4 : y;
    y = (l4 > y && y > l5) ? l5 : y;
    return y;
}

// S[b][t] = sum_{u<t} log(1 - quant(energy[b,u])), t in [0, TDIM]
__global__ void prefix_kernel(const float* __restrict__ energy,
                              float* __restrict__ S) {
    int b = threadIdx.x;
    if (b >= BSZ) return;
    const float* e = energy + (size_t)b * TDIM;
    float* s = S + (size_t)b * (TDIM + 1);
    float acc = 0.f;
    s[0] = 0.f;
    for (int t = 0; t < TDIM; ++t) {
        float x = 1.f - energy_quant(e[t]);  // in [19/32, 1], never 0
        acc += logf(x);
        s[t + 1] = acc;
    }
}

__device__ __forceinline__ float bce_logits(float z, float t) {
    return fmaxf(z, 0.f) - z * t + log1pf(__expf(-fabsf(z)));
}

// One wave32 per 16x16 lower-triangular tile. WMMA f32 16x16x4.
__global__ void __launch_bounds__(32)
tile_kernel(const float* __restrict__ Kmat,
            const float* __restrict__ XO,
            const float* __restrict__ XH,
            const float* __restrict__ S,
            const int*   __restrict__ ilens,
            float*       __restrict__ partial) {
    const int lane = threadIdx.x;
    const int half = lane >> 4;   // 0: lanes 0-15, 1: lanes 16-31
    const int l16  = lane & 15;

    // Decode (b, ti, tj) with tj <= ti from linear block index
    int idx = blockIdx.x;
    int b = idx / TILES_PER_B;
    int r = idx - b * TILES_PER_B;
    int ti = (int)((sqrtf(8.f * (float)r + 1.f) - 1.f) * 0.5f);
    while ((ti + 1) * (ti + 2) / 2 <= r) ++ti;
    while (ti * (ti + 1) / 2 > r) --ti;
    int tj = r - ti * (ti + 1) / 2;

    const int i0 = ti * 16, j0 = tj * 16;
    const int ilen = ilens[b];
    const float* Sb = S + (size_t)b * (TDIM + 1);

    // C-fragment layout: VGPR rr -> M = rr + 8*half, N = l16
    const int j = j0 + l16;            // one column per lane
    const float Sj = Sb[j];
    const bool jvalid = (j < ilen);

    float kt[8];
    bool  act[8];
    bool  laneActive = false;
#pragma unroll
    for (int rr = 0; rr < 8; ++rr) {
        int i = i0 + half * 8 + rr;
        float v;
        if (j > i)        v = 0.f;                    // strict upper -> target 0
        else if (j == i)  v = 1.f;                    // diagonal -> target 1
        else              v = __expf(Sb[i] - Sj);     // band product
        kt[rr] = v;
        bool a = jvalid && (i < ilen) && (v >= 0.1f);
        act[rr] = a;
        laneActive |= a;
    }

    // Uniform skip (EXEC stays all-ones inside, as WMMA requires)
    if (!__any(laneActive)) {
        if (lane == 0) partial[blockIdx.x] = 0.f;
        return;
    }

    // ---------- GEMM 1: ke = k @ k^T tile, K = 512 ----------
    // A fragment (16x4 f32): lanes 0-15 hold K={t,t+1}, lanes 16-31 K={t+2,t+3}
    const float* arow = Kmat + ((size_t)b * TDIM + (i0 + l16)) * HDIM;
    const float* brow = Kmat + ((size_t)b * TDIM + (j0 + l16)) * HDIM;
    v8f ce = {};
#pragma unroll 4
    for (int t = 0; t < HDIM; t += 4) {
        int o = t + 2 * half;
        v2f a, bb;
        a.x  = arow[o];  a.y  = arow[o + 1];
        bb.x = brow[o];  bb.y = brow[o + 1];
        ce = __builtin_amdgcn_wmma_f32_16x16x4_f32(
                 false, a, false, bb, (short)0, ce, false, false);
    }

    // ---------- GEMM 2: fm = xs @ xs_hat^T tile, K = 509 ----------
    const float* crow = XO + ((size_t)b * TDIM + (i0 + l16)) * DDIM;
    const float* drow = XH + ((size_t)b * TDIM + (j0 + l16)) * DDIM;
    v8f cf = {};
#pragma unroll 4
    for (int t = 0; t < DDIM - 4; t += 4) {   // full chunks: t = 0..504
        int o = t + 2 * half;
        v2f a, bb;
        a.x  = crow[o];  a.y  = crow[o + 1];
        bb.x = drow[o];  bb.y = drow[o + 1];
        cf = __builtin_amdgcn_wmma_f32_16x16x4_f32(
                 false, a, false, bb, (short)0, cf, false, false);
    }
    {   // tail chunk t = 508: only K=508 is valid (509..511 -> 0)
        int o = 508 + 2 * half;
        v2f a, bb;
        a.x  = (o     < DDIM) ? crow[o]     : 0.f;
        a.y  = (o + 1 < DDIM) ? crow[o + 1] : 0.f;
        bb.x = (o     < DDIM) ? drow[o]     : 0.f;
        bb.y = (o + 1 < DDIM) ? drow[o + 1] : 0.f;
        cf = __builtin_amdgcn_wmma_f32_16x16x4_f32(
                 false, a, false, bb, (short)0, cf, false, false);
    }

    // Masked BCE over both losses (same mask since fc_target == kernel_target)
    float sum = 0.f;
#pragma unroll
    for (int rr = 0; rr < 8; ++rr) {
        if (act[rr]) {
            float t = kt[rr];
            sum += bce_logits(ce[rr], t);
            sum += bce_logits(cf[rr], t);
        }
    }

    // wave32 reduction
    for (int off = 16; off > 0; off >>= 1)
        sum += __shfl_down(sum, off, 32);
    if (lane == 0) partial[blockIdx.x] = sum;
}

// Deterministic single-block final reduction
__global__ void reduce_kernel(const float* __restrict__ partial, int n,
                              float* __restrict__ out) {
    __shared__ float sm[256];
    float acc = 0.f;
    for (int i = threadIdx.x; i < n; i += 256) acc += partial[i];
    sm[threadIdx.x] = acc;
    __syncthreads();
    for (int s = 128; s > 0; s >>= 1) {
        if ((int)threadIdx.x < s) sm[threadIdx.x] += sm[threadIdx.x + s];
        __syncthreads();
    }
    if (threadIdx.x == 0) out[0] = sm[0];
}

extern "C" void kernel_launch(void* const* d_in, const int* in_sizes, int n_in,
                              void* d_out, int out_size, void* d_ws, size_t ws_size,
                              hipStream_t stream) {
    // setup_inputs order: q, k, xs_pad_out, xs_pad_out_hat, energy_t, ilens
    // q (d_in[0]) is dead code in the reference loss — unused.
    const float* kk = (const float*)d_in[1];
    const float* xo = (const float*)d_in[2];
    const float* xh = (const float*)d_in[3];
    const float* en = (const float*)d_in[4];
    const int*   il = (const int*)  d_in[5];
    float* out = (float*)d_out;

    // workspace layout: S [8*(1024+1)] floats, then partials [8*2080] floats
    float* S       = (float*)d_ws;
    float* partial = S + (size_t)BSZ * (TDIM + 1);
    const int nblocks = BSZ * TILES_PER_B;  // 16640

    prefix_kernel<<<1, 32, 0, stream>>>(en, S);
    tile_kernel<<<nblocks, 32, 0, stream>>>(kk, xo, xh, S, il, partial);
    reduce_kernel<<<1, 256, 0, stream>>>(partial, nblocks, out);
}